// SimpleMoE_86406152061627
// MI455X (gfx1250) — compile-verified
//
#include <hip/hip_runtime.h>

// ---------------------------------------------------------------------------
// SimpleMoE for MI455X (gfx1250): router + fused dense-masked expert FFN.
// fp32 WMMA (v_wmma_f32_16x16x4_f32) — the op is HBM-bound on 256MB of fp32
// expert weights (~14us at 23.3 TB/s), so fp32 matrix math is free and exact.
// ---------------------------------------------------------------------------

#define HID      1024
#define FFN_DIM  4096
#define NEXP     8
#define NTOK     8192          // BATCH * SEQ = 4 * 2048

#define BM       32            // tokens per workgroup
#define BF       128           // FFN chunk per iteration (16 cols per wave)
#define XSTRIDE  (HID + 4)     // LDS row stride for x  (stride % 64 banks == 4)
#define HSTRIDE  (BF + 4)      // LDS row stride for h

typedef __attribute__((ext_vector_type(2))) float v2f;
typedef __attribute__((ext_vector_type(8))) float v8f;

__device__ __forceinline__ v8f wmma_f32(v2f a, v2f b, v8f c) {
  // D = A(16x4) * B(4x16) + C  — 8-arg VOP3P pattern:
  // (neg_a, A, neg_b, B, c_mod, C, reuse_a, reuse_b)
  return __builtin_amdgcn_wmma_f32_16x16x4_f32(
      false, a, false, b, (short)0, c, false, false);
}

// ---------------------------------------------------------------------------
// Router: one wave per token. logits = x @ w_router + b, softmax, top-2,
// renormalize, write dense gates [T][8] (zeros for unselected experts).
// ---------------------------------------------------------------------------
__global__ __launch_bounds__(256) void moe_router(const float* __restrict__ x,
                                                  const float* __restrict__ wr,
                                                  const float* __restrict__ br,
                                                  float* __restrict__ gates) {
  const int lane = threadIdx.x & 31;
  const int wave = threadIdx.x >> 5;
  const int t = blockIdx.x * 8 + wave;

  float acc[NEXP];
#pragma unroll
  for (int e = 0; e < NEXP; ++e) acc[e] = 0.0f;

  const float* xr = x + (size_t)t * HID;
  for (int h = lane; h < HID; h += 32) {
    const float xv = xr[h];
    const float* wrow = wr + (size_t)h * NEXP;
#pragma unroll
    for (int e = 0; e < NEXP; ++e) acc[e] = fmaf(xv, wrow[e], acc[e]);
  }
  // wave32 butterfly reduction
#pragma unroll
  for (int off = 16; off > 0; off >>= 1) {
#pragma unroll
    for (int e = 0; e < NEXP; ++e) acc[e] += __shfl_xor(acc[e], off, 32);
  }

  if (lane == 0) {
    float l[NEXP];
    float m = -1e30f;
#pragma unroll
    for (int e = 0; e < NEXP; ++e) {
      l[e] = acc[e] + br[e];
      m = fmaxf(m, l[e]);
    }
    float p[NEXP];
#pragma unroll
    for (int e = 0; e < NEXP; ++e) p[e] = __expf(l[e] - m);
    int i0 = 0;
#pragma unroll
    for (int e = 1; e < NEXP; ++e)
      if (p[e] > p[i0]) i0 = e;
    int i1 = (i0 == 0) ? 1 : 0;
#pragma unroll
    for (int e = 0; e < NEXP; ++e)
      if (e != i0 && p[e] > p[i1]) i1 = e;
    const float s = p[i0] + p[i1];   // renormalized top-2 gate weights
    float* gr = gates + (size_t)t * NEXP;
#pragma unroll
    for (int e = 0; e < NEXP; ++e) gr[e] = 0.0f;
    gr[i0] = p[i0] / s;
    gr[i1] = p[i1] / s;
  }
}

// ---------------------------------------------------------------------------
// Fused expert FFN. One workgroup (8 waves) per 32-token block:
//   for e in experts:                       (gate==0 rows contribute 0)
//     for fc in FFN chunks of 128:
//       h   = gate[:,e] * relu(xblk @ w1[e][:,fc:fc+128] + b1)  (via LDS)
//       out += h @ w2[e][fc:fc+128, :]                          (registers)
//     out += gate[:,e] * b2[e]
// Each wave owns 128 H-output columns: 2 row-tiles x 8 col-tiles of v8f acc.
// ---------------------------------------------------------------------------
__global__ __launch_bounds__(256, 1) void moe_ffn(const float* __restrict__ x,
                                                  const float* __restrict__ gates,
                                                  const float* __restrict__ w1,
                                                  const float* __restrict__ b1,
                                                  const float* __restrict__ w2,
                                                  const float* __restrict__ b2,
                                                  float* __restrict__ out) {
  extern __shared__ float smem[];
  float* lds_x = smem;                           // BM x XSTRIDE
  float* lds_h = smem + BM * XSTRIDE;            // BM x HSTRIDE
  float* lds_g = lds_h + BM * HSTRIDE;           // BM x NEXP

  const int tid  = threadIdx.x;
  const int wave = tid >> 5;
  const int lane = tid & 31;
  const int half = lane >> 4;    // 0: K pair {0,1}, 1: K pair {2,3}
  const int lr   = lane & 15;    // row (A) / col (B,C,D) within 16x16 tile
  const int t0   = blockIdx.x * BM;
  const int col0 = wave * 128;   // this wave's H-output column base

  // Stage x block into LDS (float4, padded rows)
  for (int i = tid; i < BM * (HID / 4); i += 256) {
    const int r = i / (HID / 4);
    const int c = (i - r * (HID / 4)) * 4;
    const float4 v = *(const float4*)(x + (size_t)(t0 + r) * HID + c);
    *(float4*)&lds_x[r * XSTRIDE + c] = v;
  }
  for (int i = tid; i < BM * NEXP; i += 256)
    lds_g[i] = gates[(size_t)t0 * NEXP + i];
  __syncthreads();

  const v8f zero8 = {0.f, 0.f, 0.f, 0.f, 0.f, 0.f, 0.f, 0.f};
  v8f oacc[2][8];
#pragma unroll
  for (int rt = 0; rt < 2; ++rt)
#pragma unroll
    for (int ct = 0; ct < 8; ++ct) oacc[rt][ct] = zero8;

  for (int e = 0; e < NEXP; ++e) {
    const float* W1 = w1 + (size_t)e * HID * FFN_DIM;
    const float* W2 = w2 + (size_t)e * FFN_DIM * HID;

    // gate-weighted b2 contribution (C/D layout: VGPR j -> row j + 8*half)
#pragma unroll
    for (int ct = 0; ct < 8; ++ct) {
      const float b2v = b2[(size_t)e * HID + col0 + ct * 16 + lr];
#pragma unroll
      for (int rt = 0; rt < 2; ++rt)
#pragma unroll
        for (int j = 0; j < 8; ++j) {
          const int r = rt * 16 + half * 8 + j;
          oacc[rt][ct][j] += lds_g[r * NEXP + e] * b2v;
        }
    }

    for (int fc = 0; fc < FFN_DIM; fc += BF) {
      const int f0 = fc + wave * 16;   // this wave's 16 FFN columns

      // ---- GEMM1: h[BM x 16] = xblk @ w1[:, f0:f0+16] ----
      v8f h0 = zero8, h1 = zero8;
      const float* w1p = W1 + (size_t)f0 + lr;
      for (int k = 0; k < HID; k += 4) {
        const int kb = k + half * 2;
        v2f bmat;
        bmat.x = w1p[(size_t)kb * FFN_DIM];        // B[kb  ][lr]
        bmat.y = w1p[(size_t)(kb + 1) * FFN_DIM];  // B[kb+1][lr]
        const v2f a0 = *(const v2f*)&lds_x[lr * XSTRIDE + kb];
        const v2f a1 = *(const v2f*)&lds_x[(16 + lr) * XSTRIDE + kb];
        h0 = wmma_f32(a0, bmat, h0);
        h1 = wmma_f32(a1, bmat, h1);
      }

      // bias + relu + fold gate into h, stash in LDS for the K side of GEMM2
      const float b1v = b1[(size_t)e * FFN_DIM + f0 + lr];
#pragma unroll
      for (int j = 0; j < 8; ++j) {
        const int r0 = half * 8 + j;
        const float v0 = fmaxf(h0[j] + b1v, 0.0f) * lds_g[r0 * NEXP + e];
        lds_h[r0 * HSTRIDE + wave * 16 + lr] = v0;
        const int r1 = 16 + half * 8 + j;
        const float v1 = fmaxf(h1[j] + b1v, 0.0f) * lds_g[r1 * NEXP + e];
        lds_h[r1 * HSTRIDE + wave * 16 + lr] = v1;
      }
      __syncthreads();

      // ---- GEMM2: oacc += h[BM x BF] @ w2[fc:fc+BF, col0:col0+128] ----
      const float* w2p = W2 + (size_t)fc * HID + col0 + lr;
      for (int k = 0; k < BF; k += 4) {
        const int kb = k + half * 2;
        const v2f a0 = *(const v2f*)&lds_h[lr * HSTRIDE + kb];
        const v2f a1 = *(const v2f*)&lds_h[(16 + lr) * HSTRIDE + kb];
#pragma unroll
        for (int ct = 0; ct < 8; ++ct) {
          v2f bmat;
          bmat.x = w2p[(size_t)kb * HID + ct * 16];
          bmat.y = w2p[(size_t)(kb + 1) * HID + ct * 16];
          oacc[0][ct] = wmma_f32(a0, bmat, oacc[0][ct]);
          oacc[1][ct] = wmma_f32(a1, bmat, oacc[1][ct]);
        }
      }
      __syncthreads();
    }
  }

  // Write result (each token row written exactly once — no atomics)
#pragma unroll
  for (int rt = 0; rt < 2; ++rt)
#pragma unroll
    for (int ct = 0; ct < 8; ++ct)
#pragma unroll
      for (int j = 0; j < 8; ++j) {
        const int r = t0 + rt * 16 + half * 8 + j;
        const int c = col0 + ct * 16 + lr;
        out[(size_t)r * HID + c] = oacc[rt][ct][j];
      }
}

extern "C" void kernel_launch(void* const* d_in, const int* in_sizes, int n_in,
                              void* d_out, int out_size, void* d_ws, size_t ws_size,
                              hipStream_t stream) {
  (void)in_sizes; (void)n_in; (void)out_size; (void)ws_size;
  const float* x  = (const float*)d_in[0];
  const float* wr = (const float*)d_in[1];
  const float* br = (const float*)d_in[2];
  const float* w1 = (const float*)d_in[3];
  const float* b1 = (const float*)d_in[4];
  const float* w2 = (const float*)d_in[5];
  const float* b2 = (const float*)d_in[6];
  // d_in[7] = top_k (== 2, baked into the router kernel)
  float* out   = (float*)d_out;
  float* gates = (float*)d_ws;   // NTOK * NEXP floats = 256 KB scratch

  moe_router<<<NTOK / 8, 256, 0, stream>>>(x, wr, br, gates);

  const size_t smem_bytes =
      (size_t)(BM * XSTRIDE + BM * HSTRIDE + BM * NEXP) * sizeof(float);
  moe_ffn<<<NTOK / BM, 256, smem_bytes, stream>>>(x, gates, w1, b1, w2, b2, out);
}